// CapsuleLayer_37538014167104
// MI455X (gfx1250) — compile-verified
//
#include <hip/hip_runtime.h>

// CapsNet dynamic-routing forward, fully fused for MI455X (gfx1250, wave32).
//
// x:       [B=64, I=32, A=8, H=12, W=12] f32
// weights: [I=32, A=8, O*At=160]         f32
// bias:    [O=10, At=16]                 f32
// out:     [B=64, O=10, At=16, H=12, W=12] f32
//
// One workgroup per (b, 16-position tile). 10 waves; wave o owns output
// capsule o (16 atoms == one WMMA N-tile).
//   pass A: votes recomputed with V_WMMA_F32_16X16X4_F32 (A = x-tile,
//           B = weights, M=pos, K=atoms-in, N=atoms-out), route-scaled and
//           accumulated over input capsules in registers.
//   pass B: agreement via reassociated GEMM g = W x act^T (M = two input
//           capsules' a-rows, K = 16 output atoms, N = 16 positions), then an
//           8-term register dot with x. No lane shuffles, no divergence.

typedef __attribute__((ext_vector_type(2))) float v2f;
typedef __attribute__((ext_vector_type(8))) float v8f;

#define NUM_I   32
#define NUM_A   8
#define NUM_O   10
#define NUM_AT  16
#define POS     16
#define HW      144
#define NT      320    // 10 waves
#define WSTR    168    // w_lds row stride: 2*168 mod 64 = 16 -> halves disjoint
#define XSTR    20     // x_lds row stride: 2*20=40, 8*20=160(mod 64=32) -> clean
#define ASTR    18     // act_lds row stride: conflict-free store + B-load

// D = A(16x4) * B(4x16) + C   (fp32)
#define WMMA_F32_4(A_, B_, C_) \
  __builtin_amdgcn_wmma_f32_16x16x4_f32(false, (A_), false, (B_), (short)0, (C_), false, false)

__launch_bounds__(NT, 1)
__global__ void capsule_routing_kernel(const float* __restrict__ x,
                                       const float* __restrict__ w,
                                       const float* __restrict__ bias,
                                       float* __restrict__ out) {
  __shared__ float x_lds[NUM_I * NUM_A * XSTR];       // [i][a][p pad]   20 KB
  __shared__ float logits_lds[NUM_I * NUM_O * POS];   // [i][o][p]       20 KB
  __shared__ float route_lds[NUM_I * NUM_O * POS];    // [i][o][p]       20 KB
  __shared__ float w_lds[NUM_I * NUM_A * WSTR];       // [i][a][oat pad] 168 KB
  __shared__ float act_lds[NUM_O * NUM_AT * ASTR];    // [o][at][p pad]  11.25 KB

  const int tid  = threadIdx.x;
  const int lane = tid & 31;
  const int o    = tid >> 5;     // wave id == output capsule
  const int n    = lane & 15;    // N column (pos in D) / M row for A operands
  const int h    = lane >> 4;    // lane half (K pair for A/B, M+8 for D)

  const int b   = blockIdx.x / 9;
  const int hw0 = (blockIdx.x % 9) * POS;

  // ---- stage x tile + weights into LDS; zero logits ----
  for (int e = tid; e < NUM_I * NUM_A * POS; e += NT) {
    const int row = e >> 4;          // i*8 + a
    const int p   = e & 15;
    x_lds[row * XSTR + p] = x[((b * NUM_I) * NUM_A + row) * HW + hw0 + p];
  }
  for (int e = tid; e < NUM_I * NUM_A * (NUM_O * NUM_AT); e += NT) {
    const int row = e / (NUM_O * NUM_AT);
    const int c   = e - row * (NUM_O * NUM_AT);
    w_lds[row * WSTR + c] = w[e];
  }
  for (int e = tid; e < NUM_I * NUM_O * POS; e += NT)
    logits_lds[e] = 0.0f;

  const float bias_v = bias[o * NUM_AT + n];

  __syncthreads();

  for (int it = 0; it < 3; ++it) {
    // ---- route = softmax(logits, axis=o), per (i, pos) ----
    for (int item = tid; item < NUM_I * POS; item += NT) {
      const int i = item >> 4;
      const int p = item & 15;
      float lv[NUM_O];
      float mx = -1e30f;
#pragma unroll
      for (int oo = 0; oo < NUM_O; ++oo) {
        lv[oo] = logits_lds[(i * NUM_O + oo) * POS + p];
        mx = fmaxf(mx, lv[oo]);
      }
      float s = 0.0f;
#pragma unroll
      for (int oo = 0; oo < NUM_O; ++oo) { lv[oo] = __expf(lv[oo] - mx); s += lv[oo]; }
      const float inv = 1.0f / s;
#pragma unroll
      for (int oo = 0; oo < NUM_O; ++oo)
        route_lds[(i * NUM_O + oo) * POS + p] = lv[oo] * inv;
    }
    __syncthreads();

    // ---- pass A: preactivate[pos,o,at] = sum_i route[i,pos,o]*votes[i,pos,o,at]
    float pre[8];
#pragma unroll
    for (int r = 0; r < 8; ++r) pre[r] = 0.0f;

#pragma unroll 4
    for (int i = 0; i < NUM_I; ++i) {
      // A = x (16 pos x 4 a): element (m,k) -> lane = m + 16*(k>=2), vgpr = k&1
      v2f a0, a1, b0, b1;
      a0.x = x_lds[(i * NUM_A + 2 * h    ) * XSTR + n];
      a0.y = x_lds[(i * NUM_A + 2 * h + 1) * XSTR + n];
      a1.x = x_lds[(i * NUM_A + 4 + 2 * h    ) * XSTR + n];
      a1.y = x_lds[(i * NUM_A + 4 + 2 * h + 1) * XSTR + n];
      // B = W_i (4 a x 16 at): element (k,n) -> lane = n + 16*(k>=2), vgpr = k&1
      b0.x = w_lds[(i * NUM_A + 2 * h    ) * WSTR + o * NUM_AT + n];
      b0.y = w_lds[(i * NUM_A + 2 * h + 1) * WSTR + o * NUM_AT + n];
      b1.x = w_lds[(i * NUM_A + 4 + 2 * h    ) * WSTR + o * NUM_AT + n];
      b1.y = w_lds[(i * NUM_A + 4 + 2 * h + 1) * WSTR + o * NUM_AT + n];
      v8f d = {};
      d = WMMA_F32_4(a0, b0, d);   // a = 0..3
      d = WMMA_F32_4(a1, b1, d);   // a = 4..7
#pragma unroll
      for (int r = 0; r < 8; ++r) {
        const float rt = route_lds[(i * NUM_O + o) * POS + r + 8 * h];
        pre[r] += rt * d[r];
      }
    }

    // ---- bias + squash over atoms: act = t * sqrt(ns) / (1 + ns) ----
    float act[8];
#pragma unroll
    for (int r = 0; r < 8; ++r) {
      pre[r] += bias_v;
      float ns = pre[r] * pre[r];
      ns += __shfl_xor(ns, 1, 32);
      ns += __shfl_xor(ns, 2, 32);
      ns += __shfl_xor(ns, 4, 32);
      ns += __shfl_xor(ns, 8, 32);     // sum over 16 atoms within lane-half
      act[r] = pre[r] * (sqrtf(ns) / (1.0f + ns));
    }

    if (it < 2) {
      // stage act^T for this capsule: act_lds[o][at][pos]  (wave-local)
#pragma unroll
      for (int r = 0; r < 8; ++r)
        act_lds[(o * NUM_AT + n) * ASTR + r + 8 * h] = act[r];

      // B = act^T (4 at x 16 pos), loop-invariant over input-capsule pairs
      v2f bb[4];
#pragma unroll
      for (int kc = 0; kc < 4; ++kc) {
        bb[kc].x = act_lds[(o * NUM_AT + kc * 4 + 2 * h    ) * ASTR + n];
        bb[kc].y = act_lds[(o * NUM_AT + kc * 4 + 2 * h + 1) * ASTR + n];
      }

      // ---- pass B: g = W x act^T; dist[i,pos] = sum_a x[i,a,pos]*g[i,a,pos]
      // M rows m = a + 8*ihalf pack two input capsules (i0, i0+1).
      const int arow = (n & 7);     // a index of this lane's A row
      const int aih  = (n >> 3);    // which capsule of the pair
#pragma unroll 4
      for (int ip = 0; ip < NUM_I / 2; ++ip) {
        const int i0 = 2 * ip;
        v8f d = {};
#pragma unroll
        for (int kc = 0; kc < 4; ++kc) {
          // A = W rows: element (m, k=at) -> lane = m + 16*(k>=2), vgpr = k&1
          v2f aa;
          aa.x = w_lds[((i0 + aih) * NUM_A + arow) * WSTR + o * NUM_AT + kc * 4 + 2 * h    ];
          aa.y = w_lds[((i0 + aih) * NUM_A + arow) * WSTR + o * NUM_AT + kc * 4 + 2 * h + 1];
          d = WMMA_F32_4(aa, bb[kc], d);
        }
        // lane holds g[i0+h, a=r, pos=n] in d[r]; dot with x over a
        const int i = i0 + h;
        float dist = 0.0f;
#pragma unroll
        for (int r = 0; r < 8; ++r)
          dist += x_lds[(i * NUM_A + r) * XSTR + n] * d[r];
        logits_lds[(i * NUM_O + o) * POS + n] += dist;   // unique (i,o,pos) per lane
      }
      __syncthreads();
    } else {
      // ---- final iteration: write activation [B, O, At, H, W] ----
#pragma unroll
      for (int r = 0; r < 8; ++r)
        out[((b * NUM_O + o) * NUM_AT + n) * HW + hw0 + r + 8 * h] = act[r];
    }
  }
}

extern "C" void kernel_launch(void* const* d_in, const int* in_sizes, int n_in,
                              void* d_out, int out_size, void* d_ws, size_t ws_size,
                              hipStream_t stream) {
  const float* x    = (const float*)d_in[0];   // [64,32,8,12,12]
  const float* wts  = (const float*)d_in[1];   // [32,8,160]
  const float* bias = (const float*)d_in[2];   // [10,16,1,1]
  float* out = (float*)d_out;                  // [64,10,16,12,12]
  (void)in_sizes; (void)n_in; (void)out_size; (void)d_ws; (void)ws_size;

  dim3 grid(64 * 9);   // one workgroup per (batch, 16-position tile)
  dim3 block(NT);      // 10 waves: one per output capsule
  capsule_routing_kernel<<<grid, block, 0, stream>>>(x, wts, bias, out);
}